// GCN_68942815035652
// MI455X (gfx1250) — compile-verified
//
#include <hip/hip_runtime.h>
#include <hip/hip_bf16.h>
#include <math.h>

typedef float v2f __attribute__((ext_vector_type(2)));
typedef float v8f __attribute__((ext_vector_type(8)));

#define EPS_BN 1e-5f

// ---------------------------------------------------------------------------
// Degree / norm preparation
// ---------------------------------------------------------------------------
__global__ void gcn5_deg_init(float* deg, int n) {
    int i = blockIdx.x * blockDim.x + threadIdx.x;
    if (i < n) deg[i] = 1.0f;               // self-loop weight = 1
}

__global__ void gcn5_deg_acc(float* deg, const int* dst, const float* w, int e) {
    int i = blockIdx.x * blockDim.x + threadIdx.x;
    if (i < e) atomicAdd(&deg[dst[i]], w[i]);
}

__global__ void gcn5_dinv(float* deg, int n) {            // in-place deg -> d^-1/2
    int i = blockIdx.x * blockDim.x + threadIdx.x;
    if (i < n) { float d = deg[i]; deg[i] = (d > 0.f) ? rsqrtf(d) : 0.f; }
}

__global__ void gcn5_norm(float* norm, const int* src, const int* dst,
                          const float* w, const float* dinv, int e) {
    int i = blockIdx.x * blockDim.x + threadIdx.x;
    if (i < e) norm[i] = dinv[src[i]] * w[i] * dinv[dst[i]];
}

// ---------------------------------------------------------------------------
// WMMA fp32 GEMM:  Out[N x (NT*16)] = act(A[N x 64]) @ W[64 x (NT*16)]
// act = BN scale/shift + ReLU when BN=true (fused on the A-fragment load).
// One wave computes a 16-row strip; block = 4 waves = 64 rows.
// W is staged in LDS with K-pairs interleaved so each B fragment is one
// 8-byte-aligned ds_load_b64 straight into the WMMA operand pair:
//   sW[((k>>1)*FO + n)*2 + (k&1)] = W[k*FO + n]
// ---------------------------------------------------------------------------
template <int NT, bool BN>
__global__ __launch_bounds__(128) void gcn5_gemm_wmma(
    const float* __restrict__ A, const float* __restrict__ W,
    float* __restrict__ Out, int nn,
    const float* __restrict__ scale, const float* __restrict__ shift)
{
    constexpr int FO = NT * 16;
    __shared__ float sW[64 * FO];
    for (int i = threadIdx.x; i < 64 * FO; i += 128) {
        const int k = i / FO, n = i % FO;
        sW[((k >> 1) * FO + n) * 2 + (k & 1)] = W[i];
    }
    __syncthreads();

    const int wave = threadIdx.x >> 5;
    const int lane = threadIdx.x & 31;
    const int m0 = blockIdx.x * 64 + wave * 16;
    if (m0 >= nn) return;                    // wave-uniform (nn % 16 == 0)

    const int half = lane >> 4;              // 0: lanes 0-15, 1: lanes 16-31
    const int l    = lane & 15;
    const int m    = m0 + l;
    const float* arow = A + (size_t)m * 64;

    v8f acc[NT] = {};
#pragma unroll
    for (int k0 = 0; k0 < 64; k0 += 4) {
        // A fragment (16x4 f32): lanes 0-15 hold K=k0,k0+1; lanes 16-31 K=k0+2,k0+3
        const int ka = k0 + half * 2;
        float a0 = arow[ka], a1 = arow[ka + 1];
        if (BN) {
            a0 = fmaxf(fmaf(a0, scale[ka],     shift[ka]),     0.f);
            a1 = fmaxf(fmaf(a1, scale[ka + 1], shift[ka + 1]), 0.f);
        }
        v2f a; a.x = a0; a.y = a1;
        const int kp = (k0 >> 1) + half;     // interleaved K-pair index
#pragma unroll
        for (int t = 0; t < NT; ++t) {
            const int n = t * 16 + l;
            // B fragment (4x16 f32): rows kp*2, kp*2+1 -> one contiguous b64
            v2f b = *(const v2f*)&sW[(kp * FO + n) * 2];
            acc[t] = __builtin_amdgcn_wmma_f32_16x16x4_f32(
                false, a, false, b, (short)0, acc[t], false, false);
        }
    }
    // D layout: VGPR r -> M = m0 + r + half*8, N = l
#pragma unroll
    for (int t = 0; t < NT; ++t) {
        const int n = t * 16 + l;
#pragma unroll
        for (int r = 0; r < 8; ++r) {
            const int mm = m0 + r + half * 8;
            Out[(size_t)mm * FO + n] = acc[t][r];
        }
    }
}

// ---------------------------------------------------------------------------
// out[i][f] = bias[f] + xw[i][f] * dinv[i]^2   (bias seed + self-loop message)
// ---------------------------------------------------------------------------
__global__ void gcn5_outinit64(float* __restrict__ out, const float* __restrict__ xw,
                               const float* __restrict__ bias,
                               const float* __restrict__ dinv, int nn) {
    int idx = blockIdx.x * blockDim.x + threadIdx.x;
    if (idx >= nn * 64) return;
    int i = idx >> 6, f = idx & 63;
    float di = dinv[i];
    out[idx] = bias[f] + xw[idx] * di * di;
}

__global__ void gcn5_outinit16(float* __restrict__ out, const float* __restrict__ xw,
                               const float* __restrict__ bias,
                               const float* __restrict__ dinv, int nn) {
    int idx = blockIdx.x * blockDim.x + threadIdx.x;
    if (idx >= nn * 16) return;
    int i = idx >> 4, c = idx & 15;
    float di = dinv[i];
    out[idx] = (c < 10) ? (bias[c] + xw[idx] * di * di) : 0.f;
}

// ---------------------------------------------------------------------------
// Edge scatter:  out[dst] += xw[src] * norm   (16 lanes/edge, float4 payload)
// ---------------------------------------------------------------------------
__global__ __launch_bounds__(256) void gcn5_scatter64(
    const float* __restrict__ xw, const int* __restrict__ src,
    const int* __restrict__ dst, const float* __restrict__ norm,
    float* __restrict__ out, int e)
{
    int tid = blockIdx.x * blockDim.x + threadIdx.x;
    if (tid >= e * 16) return;
    int ed = tid >> 4, g = tid & 15;
    int s = src[ed], d = dst[ed];
    float w = norm[ed];
    float4 v = ((const float4*)(xw + (size_t)s * 64))[g];
    float* o = out + (size_t)d * 64 + g * 4;
    atomicAdd(o + 0, v.x * w);
    atomicAdd(o + 1, v.y * w);
    atomicAdd(o + 2, v.z * w);
    atomicAdd(o + 3, v.w * w);
}

__global__ __launch_bounds__(256) void gcn5_scatter16(
    const float* __restrict__ xw, const int* __restrict__ src,
    const int* __restrict__ dst, const float* __restrict__ norm,
    float* __restrict__ out, int e)
{
    int tid = blockIdx.x * blockDim.x + threadIdx.x;
    if (tid >= e * 16) return;
    int ed = tid >> 4, c = tid & 15;
    if (c >= 10) return;
    int s = src[ed], d = dst[ed];
    atomicAdd(out + (size_t)d * 16 + c, xw[(size_t)s * 16 + c] * norm[ed]);
}

// ---------------------------------------------------------------------------
// BatchNorm statistics: stats[0..63]=sum, stats[64..127]=sumsq
// ---------------------------------------------------------------------------
__global__ void gcn5_zero_stats(float* stats) {
    if (threadIdx.x < 128) stats[threadIdx.x] = 0.f;
}

__global__ __launch_bounds__(256) void gcn5_bnstats(
    const float* __restrict__ h, float* __restrict__ stats, int nn)
{
    __shared__ float ssum[64], ssq[64];
    int f = threadIdx.x & 63;
    int rl = threadIdx.x >> 6;
    if (threadIdx.x < 64) { ssum[threadIdx.x] = 0.f; ssq[threadIdx.x] = 0.f; }
    __syncthreads();
    float ls = 0.f, lq = 0.f;
    for (int r = blockIdx.x * 4 + rl; r < nn; r += gridDim.x * 4) {
        float v = h[(size_t)r * 64 + f];
        ls += v; lq += v * v;
    }
    atomicAdd(&ssum[f], ls);
    atomicAdd(&ssq[f], lq);
    __syncthreads();
    if (threadIdx.x < 64) {
        atomicAdd(&stats[threadIdx.x], ssum[threadIdx.x]);
        atomicAdd(&stats[64 + threadIdx.x], ssq[threadIdx.x]);
    }
}

// stats[128..191]=scale, stats[192..255]=shift  (affine folded)
__global__ void gcn5_bnfinal(float* stats, const float* g, const float* beta, int nn) {
    int f = threadIdx.x;
    if (f >= 64) return;
    float inv_n = 1.f / (float)nn;
    float mu  = stats[f] * inv_n;
    float var = stats[64 + f] * inv_n - mu * mu;
    float sc  = g[f] * rsqrtf(var + EPS_BN);
    stats[128 + f] = sc;
    stats[192 + f] = beta[f] - mu * sc;
}

// ---------------------------------------------------------------------------
// W3 [64x10] -> padded [64x16] (zero pad)
// ---------------------------------------------------------------------------
__global__ void gcn5_packw3(float* wp, const float* w3) {
    int idx = blockIdx.x * blockDim.x + threadIdx.x;
    if (idx >= 64 * 16) return;
    int k = idx >> 4, c = idx & 15;
    wp[idx] = (c < 10) ? w3[k * 10 + c] : 0.f;
}

// ---------------------------------------------------------------------------
// log_softmax over 10 classes (input stride 16)
// ---------------------------------------------------------------------------
__global__ void gcn5_logsoftmax(float* __restrict__ out,
                                const float* __restrict__ L, int nn) {
    int i = blockIdx.x * blockDim.x + threadIdx.x;
    if (i >= nn) return;
    const float* p = L + (size_t)i * 16;
    float mx = p[0];
#pragma unroll
    for (int c = 1; c < 10; ++c) mx = fmaxf(mx, p[c]);
    float s = 0.f;
#pragma unroll
    for (int c = 0; c < 10; ++c) s += expf(p[c] - mx);
    float ls = logf(s);
    float* o = out + (size_t)i * 10;
#pragma unroll
    for (int c = 0; c < 10; ++c) o[c] = p[c] - mx - ls;
}

// ---------------------------------------------------------------------------
extern "C" void kernel_launch(void* const* d_in, const int* in_sizes, int n_in,
                              void* d_out, int out_size, void* d_ws, size_t ws_size,
                              hipStream_t stream) {
    const int nn = in_sizes[0] / 64;     // 100000
    const int ee = in_sizes[2];          // 1600000

    const float* x    = (const float*)d_in[0];
    const int*   ei   = (const int*)d_in[1];
    const float* ew   = (const float*)d_in[2];
    const float* W1   = (const float*)d_in[3];
    const float* b1   = (const float*)d_in[4];
    const float* g1   = (const float*)d_in[5];
    const float* be1  = (const float*)d_in[6];
    const float* W2   = (const float*)d_in[7];
    const float* b2   = (const float*)d_in[8];
    const float* g2   = (const float*)d_in[9];
    const float* be2  = (const float*)d_in[10];
    const float* W3   = (const float*)d_in[11];
    const float* b3   = (const float*)d_in[12];
    const int* srcI = ei;
    const int* dstI = ei + ee;

    float* ws    = (float*)d_ws;
    float* dinv  = ws;                         // nn
    float* norm  = dinv + nn;                  // ee
    float* bufA  = norm + ee;                  // nn*64  (XW / messages)
    float* bufH  = bufA + (size_t)nn * 64;     // nn*64  (conv output)
    float* lxw   = bufH + (size_t)nn * 64;     // nn*16
    float* lout  = lxw  + (size_t)nn * 16;     // nn*16
    float* stats = lout + (size_t)nn * 16;     // 256
    float* w3pad = stats + 256;                // 64*16

    const int TB = 256;
    dim3 bN((nn + TB - 1) / TB), bE((ee + TB - 1) / TB);
    dim3 bN64(((size_t)nn * 64 + TB - 1) / TB), bN16(((size_t)nn * 16 + TB - 1) / TB);
    dim3 bE16(((size_t)ee * 16 + TB - 1) / TB);
    dim3 bG((nn + 63) / 64);

    // --- normalization coefficients -------------------------------------
    gcn5_deg_init<<<bN, TB, 0, stream>>>(dinv, nn);
    gcn5_deg_acc<<<bE, TB, 0, stream>>>(dinv, dstI, ew, ee);
    gcn5_dinv<<<bN, TB, 0, stream>>>(dinv, nn);
    gcn5_norm<<<bE, TB, 0, stream>>>(norm, srcI, dstI, ew, dinv, ee);

    // --- layer 1 ---------------------------------------------------------
    gcn5_gemm_wmma<4, false><<<bG, 128, 0, stream>>>(x, W1, bufA, nn, nullptr, nullptr);
    gcn5_outinit64<<<bN64, TB, 0, stream>>>(bufH, bufA, b1, dinv, nn);
    gcn5_scatter64<<<bE16, TB, 0, stream>>>(bufA, srcI, dstI, norm, bufH, ee);
    gcn5_zero_stats<<<1, 128, 0, stream>>>(stats);
    gcn5_bnstats<<<512, TB, 0, stream>>>(bufH, stats, nn);
    gcn5_bnfinal<<<1, 64, 0, stream>>>(stats, g1, be1, nn);

    // --- layer 2 (BN1+ReLU fused into GEMM A-load) -----------------------
    gcn5_gemm_wmma<4, true><<<bG, 128, 0, stream>>>(bufH, W2, bufA, nn, stats + 128, stats + 192);
    gcn5_outinit64<<<bN64, TB, 0, stream>>>(bufH, bufA, b2, dinv, nn);
    gcn5_scatter64<<<bE16, TB, 0, stream>>>(bufA, srcI, dstI, norm, bufH, ee);
    gcn5_zero_stats<<<1, 128, 0, stream>>>(stats);
    gcn5_bnstats<<<512, TB, 0, stream>>>(bufH, stats, nn);
    gcn5_bnfinal<<<1, 64, 0, stream>>>(stats, g2, be2, nn);

    // --- layer 3 (BN2+ReLU fused) + log_softmax --------------------------
    gcn5_packw3<<<8, 128, 0, stream>>>(w3pad, W3);
    gcn5_gemm_wmma<1, true><<<bG, 128, 0, stream>>>(bufH, w3pad, lxw, nn, stats + 128, stats + 192);
    gcn5_outinit16<<<bN16, TB, 0, stream>>>(lout, lxw, b3, dinv, nn);
    gcn5_scatter16<<<bE16, TB, 0, stream>>>(lxw, srcI, dstI, norm, lout, ee);
    gcn5_logsoftmax<<<bN, TB, 0, stream>>>((float*)d_out, lout, nn);
}